// TextureBaker_33638183862548
// MI455X (gfx1250) — compile-verified
//
#include <hip/hip_runtime.h>
#include <hip/hip_bf16.h>
#include <math.h>

// ---------------------------------------------------------------------------
// Texture baker for MI455X (gfx1250, wave32).
//  TB_prep:        per-triangle edge/attr record precompute (96 B / triangle)
//  TB_bake_staged: 1 wave per 8x4 pixel tile; triangle records streamed into
//                  LDS with double-buffered global_load_async_to_lds_b128
//                  (ASYNCcnt), ascending-index first-hit search with a
//                  wave-uniform early exit every 16 triangles.
//
// Roofline: brute force is 262k px * 8k tris ~ 2.1G tests; with uniform-random
// UV triangles E[area] ~ 11/144 so the expected first-hit index is ~13 ->
// ascending-index search with wave-uniform early exit removes ~99% of the
// work. All streamed data (~770 KB of records) is L2-resident (192 MB L2),
// so the kernel is VALU/latency bound; the CDNA5 async global->LDS DMA
// double-buffer hides all memory latency behind the test loop.
// ---------------------------------------------------------------------------

#define TRI_BLK    48                      // triangles per LDS block
#define SUB_BLK    16                      // early-exit granularity
#define REC_F      24                      // floats per record (96 B, 16B-aligned)
#define BLK_FLOATS (TRI_BLK * REC_F)       // 1152 floats
#define BLK_BYTES  (BLK_FLOATS * 4)        // 4608 B
#define CHUNKS     (BLK_BYTES / (32 * 16)) // 9 async b128 ops per wave per block

// ---- CDNA5-specific helpers ------------------------------------------------

__device__ __forceinline__ bool wave_all_hit(bool p) {
#if __has_builtin(__builtin_amdgcn_ballot_w32)
  return __builtin_amdgcn_ballot_w32(!p) == 0u;
#else
  return __all((int)p) != 0;
#endif
}

// Split dependency counters (CDNA5): async DMA tracked with ASYNCcnt,
// LDS reads with DScnt. asm keeps us portable across toolchain builtin drift.
#define WAIT_ASYNCCNT(n) asm volatile("s_wait_asynccnt %0" ::"i"(n) : "memory")
#define WAIT_DSCNT0()    asm volatile("s_wait_dscnt 0" ::: "memory")

// Issue one block (BLK_BYTES) of async global->LDS copies. Each of the 32
// lanes moves CHUNKS x 16B. ldsByte is the wave-relative LDS byte address
// (low 32 bits of the generic pointer to a __shared__ object).
__device__ __forceinline__ void issue_async_block(const float* gsrc,
                                                  unsigned ldsByte, int lane) {
  const char* g = (const char*)gsrc + lane * 16;
  unsigned    l = ldsByte + (unsigned)(lane * 16);
#pragma unroll
  for (int i = 0; i < CHUNKS; ++i) {
    asm volatile("global_load_async_to_lds_b128 %0, %1, off"
                 :: "v"(l + (unsigned)(i * 512)), "v"(g + i * 512)
                 : "memory");
  }
}

// ---- Kernel 1: per-triangle record precompute ------------------------------
// Record layout (float idx):
//  [0..3]   e0x,e0y,q0x,q0y   edge(b,c): e=(c-b), q=b
//  [4..7]   e1x,e1y,q1x,q1y   edge(c,a): e=(a-c), q=c
//  [8..11]  e2x,e2y,q2x,q2y   edge(a,b): e=(b-a), q=a
//  [12]     inv_area          (0 if invalid)
//  [13]     valid flag        (+1 valid / -1 invalid-or-padding)
//  [14..16] attr[i0].xyz  [17..19] attr[i1].xyz  [20..22] attr[i2].xyz  [23] pad
__global__ void TB_prep(const float* __restrict__ uv,
                        const int* __restrict__ faces,
                        const float* __restrict__ attr,
                        int F, int nrec, float* __restrict__ recs) {
  int t = blockIdx.x * blockDim.x + threadIdx.x;
  if (t >= nrec) return;
  float r[REC_F];
#pragma unroll
  for (int i = 0; i < REC_F; ++i) r[i] = 0.0f;
  r[13] = -1.0f;                                   // padding: never hits
  if (t < F) {
    int i0 = faces[3 * t + 0], i1 = faces[3 * t + 1], i2 = faces[3 * t + 2];
    float ax = uv[2 * i0], ay = uv[2 * i0 + 1];
    float bx = uv[2 * i1], by = uv[2 * i1 + 1];
    float cx = uv[2 * i2], cy = uv[2 * i2 + 1];
    float area  = (bx - ax) * (cy - ay) - (by - ay) * (cx - ax);
    bool  valid = fabsf(area) > 1e-9f;
    float inv   = valid ? 1.0f / area : 0.0f;
    r[0] = cx - bx; r[1] = cy - by; r[2]  = bx; r[3]  = by;
    r[4] = ax - cx; r[5] = ay - cy; r[6]  = cx; r[7]  = cy;
    r[8] = bx - ax; r[9] = by - ay; r[10] = ax; r[11] = ay;
    r[12] = inv;
    r[13] = valid ? 1.0f : -1.0f;
    r[14] = attr[3 * i0 + 0]; r[15] = attr[3 * i0 + 1]; r[16] = attr[3 * i0 + 2];
    r[17] = attr[3 * i1 + 0]; r[18] = attr[3 * i1 + 1]; r[19] = attr[3 * i1 + 2];
    r[20] = attr[3 * i2 + 0]; r[21] = attr[3 * i2 + 1]; r[22] = attr[3 * i2 + 2];
  }
  float4* dst = (float4*)(recs + (size_t)t * REC_F);
  const float4* src = (const float4*)r;
#pragma unroll
  for (int i = 0; i < REC_F / 4; ++i) dst[i] = src[i];
}

// ---- Kernel 2: staged rasterize + interpolate ------------------------------
__global__ __launch_bounds__(32)
void TB_bake_staged(const float* __restrict__ recs, int nblk, int res,
                    float* __restrict__ out) {
  __shared__ __align__(16) float lds[2][BLK_FLOATS];

  int  lane   = threadIdx.x;                  // 0..31, wave32
  int  x      = blockIdx.x * 8 + (lane & 7);  // 8x4 spatially-coherent tile
  int  y      = blockIdx.y * 4 + (lane >> 3);
  bool active = (x < res) && (y < res);
  float fr = (float)res;
  float px = ((float)x + 0.5f) / fr;
  float py = ((float)y + 0.5f) / fr;

  bool  hit = !active;                        // inactive lanes never block exit
  float o0 = 0.0f, o1 = 0.0f, o2 = 0.0f;

  unsigned ldsB0 = (unsigned)(uintptr_t)&lds[0][0];
  unsigned ldsB1 = (unsigned)(uintptr_t)&lds[1][0];

  issue_async_block(recs, ldsB0, lane);       // prime buffer 0
  int buf = 0;
  for (int blk = 0; blk < nblk; ++blk) {
    if (blk + 1 < nblk) {
      // Warm L2 two blocks ahead, fence LDS readers of the buffer we are
      // about to overwrite, kick DMA for block blk+1, then wait until only
      // those CHUNKS copies remain outstanding => block blk is in LDS.
      if (blk + 2 < nblk)
        __builtin_prefetch(recs + (size_t)(blk + 2) * BLK_FLOATS, 0, 1);
      WAIT_DSCNT0();
      issue_async_block(recs + (size_t)(blk + 1) * BLK_FLOATS,
                        buf ? ldsB0 : ldsB1, lane);
      WAIT_ASYNCCNT(CHUNKS);
    } else {
      WAIT_ASYNCCNT(0);
    }

    const float4* base = (const float4*)&lds[buf][0];
    // Three sub-blocks of 16: break-free unrolled body (lets the compiler
    // clause/schedule the ds_load_b128s densely), with the wave-uniform
    // ballot exit between sub-blocks.
    for (int sb = 0; sb < TRI_BLK / SUB_BLK; ++sb) {
#pragma unroll
      for (int tt = 0; tt < SUB_BLK; ++tt) {
        int t = sb * SUB_BLK + tt;
        float4 e0 = base[t * 6 + 0];
        float4 e1 = base[t * 6 + 1];
        float4 e2 = base[t * 6 + 2];
        float4 m  = base[t * 6 + 3];          // inv_area, valid, v0.x, v0.y
        if (!hit) {
          // Same op sequence as the reference edge()/barycentric math.
          float w0 = e0.x * (py - e0.w) - e0.y * (px - e0.z);
          float w1 = e1.x * (py - e1.w) - e1.y * (px - e1.z);
          float w2 = e2.x * (py - e2.w) - e2.y * (px - e2.z);
          float b0 = w0 * m.x, b1 = w1 * m.x, b2 = w2 * m.x;
          if (b0 >= 0.0f && b1 >= 0.0f && b2 >= 0.0f && m.y > 0.0f) {
            float4 a4 = base[t * 6 + 4];      // v0.z, v1.x, v1.y, v1.z
            float4 c4 = base[t * 6 + 5];      // v2.x, v2.y, v2.z, pad
            o0 = b0 * m.z  + b1 * a4.y + b2 * c4.x;
            o1 = b0 * m.w  + b1 * a4.z + b2 * c4.y;
            o2 = b0 * a4.x + b1 * a4.w + b2 * c4.z;
            hit = true;                       // first (lowest-index) hit wins
          }
        }
      }
      if (wave_all_hit(hit)) break;           // uniform, every 16 triangles
    }
    if (wave_all_hit(hit)) break;             // uniform early exit, block level
    buf ^= 1;
  }

  if (active) {
    size_t o = ((size_t)y * res + x) * 3;
    out[o] = o0; out[o + 1] = o1; out[o + 2] = o2;
  }
}

// ---- Fallback (only if d_ws is too small for the record table) -------------
__global__ __launch_bounds__(32)
void TB_bake_direct(const float* __restrict__ uv, const int* __restrict__ faces,
                    const float* __restrict__ attr, int F, int res,
                    float* __restrict__ out) {
  int  lane   = threadIdx.x;
  int  x      = blockIdx.x * 8 + (lane & 7);
  int  y      = blockIdx.y * 4 + (lane >> 3);
  bool active = (x < res) && (y < res);
  float fr = (float)res;
  float px = ((float)x + 0.5f) / fr, py = ((float)y + 0.5f) / fr;
  bool  hit = !active;
  float o0 = 0.0f, o1 = 0.0f, o2 = 0.0f;
  for (int t = 0; t < F; ++t) {
    int i0 = faces[3 * t + 0], i1 = faces[3 * t + 1], i2 = faces[3 * t + 2];
    float ax = uv[2 * i0], ay = uv[2 * i0 + 1];
    float bx = uv[2 * i1], by = uv[2 * i1 + 1];
    float cx = uv[2 * i2], cy = uv[2 * i2 + 1];
    float area  = (bx - ax) * (cy - ay) - (by - ay) * (cx - ax);
    bool  valid = fabsf(area) > 1e-9f;
    float inv   = valid ? 1.0f / area : 0.0f;
    if (!hit) {
      float w0 = (cx - bx) * (py - by) - (cy - by) * (px - bx);
      float w1 = (ax - cx) * (py - cy) - (ay - cy) * (px - cx);
      float w2 = (bx - ax) * (py - ay) - (by - ay) * (px - ax);
      float b0 = w0 * inv, b1 = w1 * inv, b2 = w2 * inv;
      if (b0 >= 0.0f && b1 >= 0.0f && b2 >= 0.0f && valid) {
        o0 = b0 * attr[3 * i0 + 0] + b1 * attr[3 * i1 + 0] + b2 * attr[3 * i2 + 0];
        o1 = b0 * attr[3 * i0 + 1] + b1 * attr[3 * i1 + 1] + b2 * attr[3 * i2 + 1];
        o2 = b0 * attr[3 * i0 + 2] + b1 * attr[3 * i1 + 2] + b2 * attr[3 * i2 + 2];
        hit = true;
      }
    }
    if ((t & 15) == 15 && wave_all_hit(hit)) break;
  }
  if (active) {
    size_t o = ((size_t)y * res + x) * 3;
    out[o] = o0; out[o + 1] = o1; out[o + 2] = o2;
  }
}

// ---------------------------------------------------------------------------
extern "C" void kernel_launch(void* const* d_in, const int* in_sizes, int n_in,
                              void* d_out, int out_size, void* d_ws, size_t ws_size,
                              hipStream_t stream) {
  const float* attr  = (const float*)d_in[0];
  const float* uv    = (const float*)d_in[1];
  const int*   faces = (const int*)d_in[2];
  // d_in[3] = bake_resolution lives on device (unreadable during graph
  // capture); derive it from out_size = res*res*3 instead.
  (void)n_in;
  int F   = in_sizes[2] / 3;
  int res = (int)lround(sqrt((double)out_size / 3.0));
  float* out = (float*)d_out;

  int    nblk = (F + TRI_BLK - 1) / TRI_BLK;
  int    nrec = nblk * TRI_BLK;
  size_t need = (size_t)nrec * REC_F * sizeof(float);

  dim3 grid((res + 7) / 8, (res + 3) / 4);
  if (ws_size >= need && nblk > 0) {
    float* recs = (float*)d_ws;
    TB_prep<<<(nrec + 255) / 256, 256, 0, stream>>>(uv, faces, attr, F, nrec, recs);
    TB_bake_staged<<<grid, 32, 0, stream>>>(recs, nblk, res, out);
  } else {
    TB_bake_direct<<<grid, 32, 0, stream>>>(uv, faces, attr, F, res, out);
  }
}